// SymmetricContraction_10428180595107
// MI455X (gfx1250) — compile-verified
//
#include <hip/hip_runtime.h>
#include <math.h>

// Problem constants (match reference)
#define NN 1024
#define CC 128
#define DD 16
#define SS 10

typedef float v2f __attribute__((ext_vector_type(2)));
typedef float v8f __attribute__((ext_vector_type(8)));

// ---------------------------------------------------------------------------
// Kernel 0: bucket nodes by species (deterministic, single-thread — 1024 elems)
// ---------------------------------------------------------------------------
__global__ void sc_sort_kernel(const int* __restrict__ y,
                               int* __restrict__ sorted,
                               int* __restrict__ soff) {
  if (threadIdx.x != 0 || blockIdx.x != 0) return;
  int cnt[SS + 1];
  for (int s = 0; s <= SS; ++s) cnt[s] = 0;
  for (int n = 0; n < NN; ++n) cnt[y[n] + 1]++;
  for (int s = 0; s < SS; ++s) cnt[s + 1] += cnt[s];
  for (int s = 0; s <= SS; ++s) soff[s] = cnt[s];
  int pos[SS];
  for (int s = 0; s < SS; ++s) pos[s] = cnt[s];
  for (int n = 0; n < NN; ++n) { int s = y[n]; sorted[pos[s]++] = n; }
}

// ---------------------------------------------------------------------------
// Kernel 1: build per-(species,channel) weight tables
//   outT[((s*C+c)*icnt + il)*Dp + e] = scale * Sum_k U[(e*k+kk)*icnt+il] * w[(s*k+kk)*C+c]
//   (U is (D^p, k, i) row-major; w is (S, k, C) row-major)
// ---------------------------------------------------------------------------
__global__ void sc_table_kernel(const float* __restrict__ U,
                                const float* __restrict__ w,
                                float* __restrict__ outT,
                                int Dp, int k, int icnt, float scale, int total) {
  int idx = blockIdx.x * blockDim.x + threadIdx.x;
  if (idx >= total) return;
  int e  = idx % Dp;
  int r0 = idx / Dp;
  int il = r0 % icnt;
  int sc = r0 / icnt;
  int c  = sc % CC;
  int s  = sc / CC;
  float acc = 0.f;
  for (int kk = 0; kk < k; ++kk)
    acc = fmaf(U[(e * k + kk) * icnt + il], w[(s * k + kk) * CC + c], acc);
  outT[idx] = acc * scale;
}

// ---------------------------------------------------------------------------
// Kernel 2: main contraction.
//   grid = (C=128, S=10), block = 256 threads = 8 waves.
//   wave -> (combo = output component 0..3, half = a-range).  All W3/W2/W1
//   fragments for this wave are loop-invariant across node chunks, so they are
//   hoisted into registers (64+64+8 VGPRs) before the chunk loop; the inner
//   loop is then pure LDS + WMMA + VALU with no global loads.
//   Per chunk (16 nodes of species s):
//     acc(16x16)[b,n'] = W2[a,b] + Sum_j W3[a,b,j] * X[j,n']  (4x WMMA f32 16x16x4)
//     partial(n')      = Sum_b acc[b,n'] * X[b,n']            (VALU + shfl_xor 16)
//     out(n')         += X[a,n'] * (partial + W1[a])
//   Halves combined deterministically through LDS, stored to d_out.
// ---------------------------------------------------------------------------
__global__ __launch_bounds__(256) void sc_contract_kernel(
    const float* __restrict__ x,
    const int*   __restrict__ sorted,
    const int*   __restrict__ soff,
    const float* __restrict__ W3_0, const float* __restrict__ W3_1,
    const float* __restrict__ W2_0, const float* __restrict__ W2_1,
    const float* __restrict__ W1_0, const float* __restrict__ W1_1,
    float* __restrict__ out) {
  __shared__ float Xlds[DD * 16];        // [j][n']
  __shared__ float outbuf[4][2][16];     // [component][a-half][n']

  const int c    = blockIdx.x;
  const int s    = blockIdx.y;
  const int t    = threadIdx.x;
  const int wave = t >> 5;
  const int lane = t & 31;
  const int hi   = lane >> 4;            // lane half (0/1)
  const int lp   = lane & 15;            // n' column
  const int combo = wave >> 1;           // 0 -> 0e(i=0); 1..3 -> 1o(i=0..2)
  const int half  = wave & 1;            // a in [half*8, half*8+8)

  const float* W3 = (combo == 0) ? (W3_0 + (size_t)(s * CC + c) * 4096)
                                 : (W3_1 + ((size_t)(s * CC + c) * 3 + (combo - 1)) * 4096);
  const float* W2 = (combo == 0) ? (W2_0 + (size_t)(s * CC + c) * 256)
                                 : (W2_1 + ((size_t)(s * CC + c) * 3 + (combo - 1)) * 256);
  const float* W1 = (combo == 0) ? (W1_0 + (size_t)(s * CC + c) * 16)
                                 : (W1_1 + ((size_t)(s * CC + c) * 3 + (combo - 1)) * 16);

  // ---- Hoist loop-invariant weight fragments into registers ----
  // A (16x4 f32) fragment layout: lane L holds row M=L%16; VGPR0/1 -> K = 4q+2*hi+{0,1}
  v2f   Afrag[8][4];
  float W2f[8][8];
  float W1f[8];
#pragma unroll
  for (int aa = 0; aa < 8; ++aa) {
    const int a = half * 8 + aa;
    const float* tile = W3 + a * 256;    // [b][j] 16x16 row-major
#pragma unroll
    for (int q = 0; q < 4; ++q) {
      Afrag[aa][q].x = tile[lp * 16 + 4 * q + 2 * hi + 0];
      Afrag[aa][q].y = tile[lp * 16 + 4 * q + 2 * hi + 1];
    }
    // C/D layout: lane L, VGPR r holds (M = r + 8*(L>=16), N = L%16)
#pragma unroll
    for (int r = 0; r < 8; ++r) W2f[aa][r] = W2[a * 16 + r + 8 * hi];
    W1f[aa] = W1[a];
  }

  const int beg = soff[s], end = soff[s + 1];

  for (int base = beg; base < end; base += 16) {
    // Cooperative load of X tile (transposed to [j][n']), zero-padded tail.
    {
      const int nl = t >> 4;   // node-local 0..15
      const int j  = t & 15;
      float v = 0.f;
      if (base + nl < end) {
        const int node = sorted[base + nl];
        v = x[((size_t)node * CC + c) * DD + j];
      }
      Xlds[j * 16 + nl] = v;
      // Prefetch next chunk's x row (only remaining global stream in the loop).
      if (base + 16 + nl < end) {
        const int nnode = sorted[base + 16 + nl];
        __builtin_prefetch(&x[((size_t)nnode * CC + c) * DD + j], 0, 1);
      }
    }
    __syncthreads();

    float outacc = 0.f;
#pragma unroll
    for (int aa = 0; aa < 8; ++aa) {
      const int a = half * 8 + aa;

      v8f acc;
#pragma unroll
      for (int r = 0; r < 8; ++r) acc[r] = W2f[aa][r];

#pragma unroll
      for (int q = 0; q < 4; ++q) {
        // B (4x16 f32): lane L holds col N=L%16; VGPR0/1 -> K = 4q + 2*hi + {0,1}
        v2f B;
        B.x = Xlds[(4 * q + 2 * hi + 0) * 16 + lp];
        B.y = Xlds[(4 * q + 2 * hi + 1) * 16 + lp];
        acc = __builtin_amdgcn_wmma_f32_16x16x4_f32(
            /*neg_a=*/false, Afrag[aa][q], /*neg_b=*/false, B,
            /*c_mod=*/(short)0, acc, /*reuse_a=*/false, /*reuse_b=*/false);
      }

      // Epilogue: Sum_b (T1 + W2)[b,n'] * x_b, halves combined via xor-16 shuffle
      float partial = 0.f;
#pragma unroll
      for (int r = 0; r < 8; ++r)
        partial = fmaf(acc[r], Xlds[(r + 8 * hi) * 16 + lp], partial);
      partial += __shfl_xor(partial, 16, 32);

      outacc = fmaf(Xlds[a * 16 + lp], partial + W1f[aa], outacc);
    }

    if (hi == 0) outbuf[combo][half][lp] = outacc;
    __syncthreads();

    if (t < 64) {
      const int cb = t >> 4, np = t & 15;
      if (base + np < end) {
        const int node = sorted[base + np];
        out[((size_t)node * CC + c) * 4 + cb] =
            outbuf[cb][0][np] + outbuf[cb][1][np];
      }
    }
    __syncthreads();
  }
}

// ---------------------------------------------------------------------------
// Workspace layout (floats). Total ~22.37M floats ~= 85.3 MB (assumed <= ws_size).
// ---------------------------------------------------------------------------
static const size_t OFF_W3_0 = 0;              // 10*128*1*4096 = 5,242,880
static const size_t OFF_W3_1 = 5242880;        // 10*128*3*4096 = 15,728,640
static const size_t OFF_W2_0 = 20971520;       // 10*128*1*256  =   327,680
static const size_t OFF_W2_1 = 21299200;       // 10*128*3*256  =   983,040
static const size_t OFF_W1_0 = 22282240;       // 10*128*1*16   =    20,480
static const size_t OFF_W1_1 = 22302720;       // 10*128*3*16   =    61,440
static const size_t OFF_SORT = 22364160;       // 1024 ints + 11 ints

extern "C" void kernel_launch(void* const* d_in, const int* in_sizes, int n_in,
                              void* d_out, int out_size, void* d_ws, size_t ws_size,
                              hipStream_t stream) {
  (void)in_sizes; (void)n_in; (void)out_size; (void)ws_size;

  const float* x    = (const float*)d_in[0];
  const int*   y    = (const int*)d_in[1];
  const float* U3_0 = (const float*)d_in[2];
  const float* w3_0 = (const float*)d_in[3];
  const float* U2_0 = (const float*)d_in[4];
  const float* w2_0 = (const float*)d_in[5];
  const float* U1_0 = (const float*)d_in[6];
  const float* w1_0 = (const float*)d_in[7];
  const float* U3_1 = (const float*)d_in[8];
  const float* w3_1 = (const float*)d_in[9];
  const float* U2_1 = (const float*)d_in[10];
  const float* w2_1 = (const float*)d_in[11];
  const float* U1_1 = (const float*)d_in[12];
  const float* w1_1 = (const float*)d_in[13];
  float* out = (float*)d_out;

  float* ws   = (float*)d_ws;
  float* W3_0 = ws + OFF_W3_0;
  float* W3_1 = ws + OFF_W3_1;
  float* W2_0 = ws + OFF_W2_0;
  float* W2_1 = ws + OFF_W2_1;
  float* W1_0 = ws + OFF_W1_0;
  float* W1_1 = ws + OFF_W1_1;
  int*   sorted = (int*)(ws + OFF_SORT);
  int*   soff   = sorted + NN;

  // 0) species bucketing
  sc_sort_kernel<<<1, 32, 0, stream>>>(y, sorted, soff);

  // 1) tables: fold path norms (k^-1/2) and the 1/3, 1/2 einsum factors in.
  const float s3_0 = (1.0f / 3.0f) / sqrtf(23.0f);
  const float s2_0 = 0.5f / sqrtf(7.0f);
  const float s1_0 = 1.0f / sqrtf(2.0f);
  const float s3_1 = (1.0f / 3.0f) / sqrtf(17.0f);
  const float s2_1 = 0.5f / sqrtf(5.0f);
  const float s1_1 = 1.0f;  // k1 = 1

  const int t3_0 = SS * CC * 1 * 4096, t3_1 = SS * CC * 3 * 4096;
  const int t2_0 = SS * CC * 1 * 256,  t2_1 = SS * CC * 3 * 256;
  const int t1_0 = SS * CC * 1 * 16,   t1_1 = SS * CC * 3 * 16;

  sc_table_kernel<<<(t3_0 + 255) / 256, 256, 0, stream>>>(U3_0, w3_0, W3_0, 4096, 23, 1, s3_0, t3_0);
  sc_table_kernel<<<(t3_1 + 255) / 256, 256, 0, stream>>>(U3_1, w3_1, W3_1, 4096, 17, 3, s3_1, t3_1);
  sc_table_kernel<<<(t2_0 + 255) / 256, 256, 0, stream>>>(U2_0, w2_0, W2_0, 256,  7,  1, s2_0, t2_0);
  sc_table_kernel<<<(t2_1 + 255) / 256, 256, 0, stream>>>(U2_1, w2_1, W2_1, 256,  5,  3, s2_1, t2_1);
  sc_table_kernel<<<(t1_0 + 255) / 256, 256, 0, stream>>>(U1_0, w1_0, W1_0, 16,   2,  1, s1_0, t1_0);
  sc_table_kernel<<<(t1_1 + 255) / 256, 256, 0, stream>>>(U1_1, w1_1, W1_1, 16,   1,  3, s1_1, t1_1);

  // 2) main WMMA contraction
  dim3 grid(CC, SS);
  sc_contract_kernel<<<grid, 256, 0, stream>>>(x, sorted, soff,
                                               W3_0, W3_1, W2_0, W2_1, W1_0, W1_1,
                                               out);
}